// ForwardFunctionalMap_48713519071692
// MI455X (gfx1250) — compile-verified
//
#include <hip/hip_runtime.h>

// ---------------------------------------------------------------------------
// ForwardFunctionalMap on MI455X (gfx1250)
//   fmap rows solved by batched CG; operator = 512x512x512 fp32 WMMA GEMM.
// ---------------------------------------------------------------------------

#define D_DIM   1024
#define K_DIM   512
#define LMBDA_F 1000.0f
#define NITER   96

typedef float v2f __attribute__((ext_vector_type(2)));
typedef float v8f __attribute__((ext_vector_type(8)));

__device__ __forceinline__ v8f wmma_f32(v2f a, v2f b, v8f c) {
    // D = A(16x4 f32) * B(4x16 f32) + C(16x16 f32)
    // 8-arg pattern: (neg_a, A, neg_b, B, c_mod, C, reuse_a, reuse_b)
    return __builtin_amdgcn_wmma_f32_16x16x4_f32(
        false, a, false, b, (short)0, c, false, false);
}

// ---------------------------------------------------------------------------
// s = max(max(evals_a), max(evals_b));  Ka == Kb == 512, one block of 512.
// ---------------------------------------------------------------------------
__global__ void max_reduce_kernel(const float* __restrict__ ea,
                                  const float* __restrict__ eb,
                                  float* __restrict__ s_out) {
    __shared__ float sm[512];
    int t = threadIdx.x;
    sm[t] = fmaxf(ea[t], eb[t]);
    __syncthreads();
    for (int off = 256; off > 0; off >>= 1) {
        if (t < off) sm[t] = fmaxf(sm[t], sm[t + off]);
        __syncthreads();
    }
    if (t == 0) s_out[0] = sm[0];
}

// ---------------------------------------------------------------------------
// mask[i,j] from row spectrum (erow) vs col spectrum (ecol), gamma == 1.
// grid = (K/256, K), block = 256
// ---------------------------------------------------------------------------
__global__ void mask_kernel(const float* __restrict__ erow,
                            const float* __restrict__ ecol,
                            const float* __restrict__ s_ptr,
                            float* __restrict__ mask) {
    int j = blockIdx.x * blockDim.x + threadIdx.x;   // column
    int i = blockIdx.y;                              // row
    float inv_s = 1.0f / s_ptr[0];
    float gb = erow[i] * inv_s;          // gamma == 1.0 -> plain scale
    float ga = ecol[j] * inv_s;
    float gb2 = gb * gb + 1.0f;
    float ga2 = ga * ga + 1.0f;
    float re = gb / gb2 - ga / ga2;
    float im = 1.0f / gb2 - 1.0f / ga2;
    mask[i * K_DIM + j] = re * re + im * im;
}

// ---------------------------------------------------------------------------
// C[KxK] = X^T * Y,  X,Y: [D,K] row-major.
// grid = (8, 2): blockIdx.x -> 64-row M tile, blockIdx.y -> 256-wide N half.
// 8 waves/block, each wave owns a 64x32 output tile:
//   4 (M) x 2 (N) = 8 f32 accumulators (64 VGPRs) -> no spills.
// ---------------------------------------------------------------------------
__global__ void __launch_bounds__(256, 1)
gram_kernel(const float* __restrict__ X,
            const float* __restrict__ Y,
            float* __restrict__ C) {
    int lane = threadIdx.x & 31;
    int wave = threadIdx.x >> 5;
    int lo = lane & 15;        // M / N within 16x16 tile
    int hi = lane >> 4;        // selects K pair (A/B) / M+8 half (C)
    int m0 = blockIdx.x * 64;
    int n0 = blockIdx.y * 256 + wave * 32;

    v8f zero = {};
    v8f acc[4][2];
#pragma unroll
    for (int mi = 0; mi < 4; ++mi)
#pragma unroll
        for (int ni = 0; ni < 2; ++ni) acc[mi][ni] = zero;

    for (int d = 0; d < D_DIM; d += 4) {
        int kk = d + 2 * hi;
        v2f aF[4], bF[2];
#pragma unroll
        for (int mi = 0; mi < 4; ++mi) {
            int m = m0 + mi * 16 + lo;
            aF[mi].x = X[kk * K_DIM + m];
            aF[mi].y = X[(kk + 1) * K_DIM + m];
        }
#pragma unroll
        for (int ni = 0; ni < 2; ++ni) {
            int n = n0 + ni * 16 + lo;
            bF[ni].x = Y[kk * K_DIM + n];
            bF[ni].y = Y[(kk + 1) * K_DIM + n];
        }
#pragma unroll
        for (int mi = 0; mi < 4; ++mi)
#pragma unroll
            for (int ni = 0; ni < 2; ++ni)
                acc[mi][ni] = wmma_f32(aF[mi], bF[ni], acc[mi][ni]);
    }

    // C/D layout: VGPR v -> M = v + 8*hi, N = lo
#pragma unroll
    for (int mi = 0; mi < 4; ++mi)
#pragma unroll
        for (int ni = 0; ni < 2; ++ni)
#pragma unroll
            for (int v = 0; v < 8; ++v) {
                int m = m0 + mi * 16 + v + 8 * hi;
                int n = n0 + ni * 16 + lo;
                C[m * K_DIM + n] = acc[mi][ni][v];
            }
}

// ---------------------------------------------------------------------------
// Fused batched CG. One workgroup per 16 systems (rows of the fmap).
//   system i:  (G + lambda*diag(mask[i])) x_i = R[i,:]
// Operator applied to all 16 rows at once:
//   Ap[16x512] = P[16x512] * G[512x512]  (WMMA, A-frag from LDS P)
//              + lambda * mask o P
// Dynamic LDS: p | r | x | Ap  (4 * 16*512 floats = 128 KB, < 320 KB/WGP).
// block = 256 (8 waves); each wave owns a 64-wide N strip of the GEMM
// (4 accumulators = 32 VGPRs, spill-free).
// ---------------------------------------------------------------------------
__global__ void __launch_bounds__(256, 1)
cg_solve_kernel(const float* __restrict__ G,
                const float* __restrict__ R,
                const float* __restrict__ mask,
                float* __restrict__ Xout,
                int niter) {
    extern __shared__ float smem[];
    float* p_s  = smem;                    // 16*512
    float* r_s  = p_s  + 16 * K_DIM;       // 16*512
    float* x_s  = r_s  + 16 * K_DIM;       // 16*512
    float* ap_s = x_s  + 16 * K_DIM;       // 16*512
    __shared__ float rs[16], rs_new[16], pAp[16], alpha_s[16], beta_s[16];

    int t    = threadIdx.x;
    int lane = t & 31;
    int wave = t >> 5;
    int lo   = lane & 15;
    int hi   = lane >> 4;
    int n0   = wave * 64;
    int m0   = blockIdx.x * 16;            // global first row of this WG

    int rowT = t >> 4;                     // 0..15 : row owned for reductions
    int colT = t & 15;                     // 16 threads per row, stride-16 cols

    // ---- init: r = R, p = r, x = 0, rs = <r,r> ----
    if (t < 16) rs[t] = 0.0f;
    __syncthreads();
    {
        float part = 0.0f;
        for (int j = 0; j < 32; ++j) {
            int c = colT + j * 16;
            float rv = R[(m0 + rowT) * K_DIM + c];
            r_s[rowT * K_DIM + c] = rv;
            p_s[rowT * K_DIM + c] = rv;
            x_s[rowT * K_DIM + c] = 0.0f;
            part += rv * rv;
        }
        atomicAdd(&rs[rowT], part);
    }
    __syncthreads();

    for (int it = 0; it < niter; ++it) {
        if (t < 16) pAp[t] = 0.0f;
        __syncthreads();

        // ---- Ap = p * G  (wave strip: N in [n0, n0+64)) ----
        v8f zero = {};
        v8f acc[4];
#pragma unroll
        for (int ni = 0; ni < 4; ++ni) acc[ni] = zero;

        for (int kb = 0; kb < K_DIM; kb += 4) {
            int kk = kb + 2 * hi;
            v2f aF;                                   // A-frag: rows = lo
            aF.x = p_s[lo * K_DIM + kk];
            aF.y = p_s[lo * K_DIM + kk + 1];
#pragma unroll
            for (int ni = 0; ni < 4; ++ni) {
                int n = n0 + ni * 16 + lo;
                v2f bF;
                bF.x = G[kk * K_DIM + n];
                bF.y = G[(kk + 1) * K_DIM + n];
                acc[ni] = wmma_f32(aF, bF, acc[ni]);
            }
        }

        // ---- epilogue: + lambda*mask o p ; store Ap ; accumulate pAp ----
#pragma unroll
        for (int ni = 0; ni < 4; ++ni) {
            int n = n0 + ni * 16 + lo;
#pragma unroll
            for (int v = 0; v < 8; ++v) {
                int m = v + 8 * hi;                   // local row 0..15
                float pv  = p_s[m * K_DIM + n];
                float val = acc[ni][v]
                          + LMBDA_F * mask[(m0 + m) * K_DIM + n] * pv;
                ap_s[m * K_DIM + n] = val;
                atomicAdd(&pAp[m], pv * val);
            }
        }
        __syncthreads();

        if (t < 16) { alpha_s[t] = rs[t] / pAp[t]; rs_new[t] = 0.0f; }
        __syncthreads();

        // ---- x += a*p ; r -= a*Ap ; rs_new = <r,r> ----
        {
            float a_m  = alpha_s[rowT];
            float part = 0.0f;
            for (int j = 0; j < 32; ++j) {
                int idx = rowT * K_DIM + colT + j * 16;
                float xv = x_s[idx] + a_m * p_s[idx];
                float rv = r_s[idx] - a_m * ap_s[idx];
                x_s[idx] = xv;
                r_s[idx] = rv;
                part += rv * rv;
            }
            atomicAdd(&rs_new[rowT], part);
        }
        __syncthreads();

        if (t < 16) { beta_s[t] = rs_new[t] / rs[t]; rs[t] = rs_new[t]; }
        __syncthreads();

        // ---- p = r + b*p ----
        {
            float b_m = beta_s[rowT];
            for (int j = 0; j < 32; ++j) {
                int idx = rowT * K_DIM + colT + j * 16;
                p_s[idx] = r_s[idx] + b_m * p_s[idx];
            }
        }
        __syncthreads();
    }

    // ---- write solution rows ----
    for (int j = 0; j < 32; ++j) {
        int c = colT + j * 16;
        Xout[(m0 + rowT) * K_DIM + c] = x_s[rowT * K_DIM + c];
    }
}

// ---------------------------------------------------------------------------
extern "C" void kernel_launch(void* const* d_in, const int* in_sizes, int n_in,
                              void* d_out, int out_size, void* d_ws, size_t ws_size,
                              hipStream_t stream) {
    const float* A  = (const float*)d_in[0];   // sdescr_a [D, Ka]
    const float* B  = (const float*)d_in[1];   // sdescr_b [D, Kb]
    const float* ea = (const float*)d_in[2];   // evals_a [Ka]
    const float* eb = (const float*)d_in[3];   // evals_b [Kb]
    float* out = (float*)d_out;                // fmap_12 | fmap_21

    float* ws      = (float*)d_ws;
    float* s_max   = ws;                                   // 16 (padded)
    float* mask_ab = s_max   + 16;                         // [Kb,Ka]
    float* mask_ba = mask_ab + K_DIM * K_DIM;              // [Ka,Kb]
    float* G_a     = mask_ba + K_DIM * K_DIM;              // A^T A
    float* G_b     = G_a     + K_DIM * K_DIM;              // B^T B
    float* R_ab    = G_b     + K_DIM * K_DIM;              // B^T A
    float* R_ba    = R_ab    + K_DIM * K_DIM;              // A^T B

    max_reduce_kernel<<<1, 512, 0, stream>>>(ea, eb, s_max);

    dim3 mgrid(K_DIM / 256, K_DIM);
    mask_kernel<<<mgrid, 256, 0, stream>>>(eb, ea, s_max, mask_ab); // rows = b
    mask_kernel<<<mgrid, 256, 0, stream>>>(ea, eb, s_max, mask_ba); // rows = a

    dim3 ggrid(8, 2);
    gram_kernel<<<ggrid, 256, 0, stream>>>(A, A, G_a);   // A^T A
    gram_kernel<<<ggrid, 256, 0, stream>>>(B, A, R_ab);  // B^T A
    gram_kernel<<<ggrid, 256, 0, stream>>>(B, B, G_b);   // B^T B
    gram_kernel<<<ggrid, 256, 0, stream>>>(A, B, R_ba);  // A^T B

    size_t shmem = (size_t)4 * 16 * K_DIM * sizeof(float);   // 128 KB
    cg_solve_kernel<<<K_DIM / 16, 256, shmem, stream>>>(
        G_a, R_ab, mask_ab, out, NITER);                     // fmap_12
    cg_solve_kernel<<<K_DIM / 16, 256, shmem, stream>>>(
        G_b, R_ba, mask_ba, out + K_DIM * K_DIM, NITER);     // fmap_21
}